// StackMambaBlock_49237505082086
// MI455X (gfx1250) — compile-verified
//
#include <hip/hip_runtime.h>
#include <math.h>

// ---------------------------------------------------------------------------
// StackMambaBlock for MI455X (gfx1250): f32 WMMA GEMMs (LDS-staged W tiles)
// + per-(b,h) sequential SSM scan.
// ---------------------------------------------------------------------------

typedef __attribute__((ext_vector_type(2))) float v2f;
typedef __attribute__((ext_vector_type(8))) float v8f;

#define DM      512     // d_model
#define NSTATE  64      // D_STATE
#define DINNER  1024    // EXPAND * d_model
#define NHEADS  8       // d_inner / HEADDIM
#define PDIM    128     // HEADDIM
#define CONVD   1152    // d_inner + 2*N
#define ZROWS   2184    // 2*d_inner + 2*N + H
#define LN_EPS  1e-5f

// ---------------- helpers ----------------

__device__ __forceinline__ float blockReduceSum256(float v, float* red) {
  int tid = threadIdx.x;
  red[tid] = v;
  __syncthreads();
  #pragma unroll
  for (int off = 128; off > 0; off >>= 1) {
    if (tid < off) red[tid] += red[tid + off];
    __syncthreads();
  }
  float r = red[0];
  __syncthreads();
  return r;
}

__device__ __forceinline__ float gelu_exact(float x) {
  return 0.5f * x * (1.f + erff(x * 0.70710678118654752f));
}

__device__ __forceinline__ float silu(float x) {
  return x / (1.f + expf(-x));
}

// ---------------- layernorm over D=512, batch-strided in/out ----------------

__global__ void ln_rows(const float* __restrict__ in, const float* __restrict__ g,
                        const float* __restrict__ bb, float* __restrict__ out,
                        size_t inBatch, size_t outBatch) {
  __shared__ float red[256];
  int r = blockIdx.x, b = blockIdx.y, t = threadIdx.x;
  const float* x = in + (size_t)b * inBatch + (size_t)r * DM;
  float* o = out + (size_t)b * outBatch + (size_t)r * DM;
  float x0 = x[t], x1 = x[t + 256];
  float mu = blockReduceSum256(x0 + x1, red) * (1.f / 512.f);
  float d0 = x0 - mu, d1 = x1 - mu;
  float var = blockReduceSum256(d0 * d0 + d1 * d1, red) * (1.f / 512.f);
  float rs = rsqrtf(var + LN_EPS);
  o[t]       = d0 * rs * g[t]       + bb[t];
  o[t + 256] = d1 * rs * g[t + 256] + bb[t + 256];
}

// ---------------- h = x_extra[:, :, None] * emb_w; layernorm(h) ----------------

__global__ void emb_h_ln(const float* __restrict__ xe, const float* __restrict__ ew,
                         const float* __restrict__ g, const float* __restrict__ bb,
                         float* __restrict__ out, int E) {
  __shared__ float red[256];
  int e = blockIdx.x, b = blockIdx.y, t = threadIdx.x;
  float c = xe[b * E + e];
  float x0 = c * ew[t], x1 = c * ew[t + 256];
  float mu = blockReduceSum256(x0 + x1, red) * (1.f / 512.f);
  float d0 = x0 - mu, d1 = x1 - mu;
  float var = blockReduceSum256(d0 * d0 + d1 * d1, red) * (1.f / 512.f);
  float rs = rsqrtf(var + LN_EPS);
  float* o = out + ((size_t)(b * E + e)) * DM;
  o[t]       = d0 * rs * g[t]       + bb[t];
  o[t + 256] = d1 * rs * g[t + 256] + bb[t + 256];
}

// ---------------- f32 WMMA GEMM: C[m,n] = act(sum_k A[m,k]*W[n,k] + bias) + res
// One wave per 16x16 tile, 8 waves/block sharing one column tile; the 16xK W
// slab is staged through LDS in 16x64 chunks (coalesced b128 cooperative load,
// then conflict-spread ds_load b64 fragment reads). A rows come straight from
// global (per-lane b64) with next-slab prefetch. V_WMMA_F32_16X16X4_F32.
// Output rows remapped (m -> (m/rowsPerBatch)*outRowsPerBatch + m%rowsPerBatch)
// so the emb-mamba output can land batch-strided inside the concat buffer. ----

__global__ void gemm_wmma_f32(const float* __restrict__ A, const float* __restrict__ W,
                              const float* __restrict__ bias, const float* __restrict__ res,
                              float* __restrict__ C, int R, int K, int Nout, int act,
                              int rowsPerBatch, int outRowsPerBatch) {
  __shared__ float sW[16][65];                    // 65-float row pitch spreads banks
  const int tid  = threadIdx.x;
  const int lane = tid & 31;
  const int wave = tid >> 5;
  const int rt = blockIdx.y * 8 + wave;
  const int ct = blockIdx.x;
  const bool active = (rt * 16 < R);              // wave-uniform; EXEC all-ones at WMMA
  const int half = lane >> 4;                     // 0: lanes 0-15, 1: lanes 16-31
  const int l16 = lane & 15;

  // cooperative W-chunk load mapping: thread -> (row, 4-col group) of 16x64 chunk
  const int lr = tid >> 4;                        // 0..15
  const int lc = (tid & 15) * 4;                  // 0,4,...,60
  int wrow = ct * 16 + lr;
  if (wrow >= Nout) wrow = Nout - 1;              // clamp loads for partial tile
  const float* wsrc = W + (size_t)wrow * K + lc;

  const float* arow = A + (size_t)(rt * 16 + l16) * K + 2 * half;

  v8f acc = {0.f, 0.f, 0.f, 0.f, 0.f, 0.f, 0.f, 0.f};
  for (int kc = 0; kc < K; kc += 64) {
    float4 wv = *(const float4*)(wsrc + kc);      // global_load_b128, coalesced
    __syncthreads();                              // previous chunk fully consumed
    sW[lr][lc + 0] = wv.x;
    sW[lr][lc + 1] = wv.y;
    sW[lr][lc + 2] = wv.z;
    sW[lr][lc + 3] = wv.w;
    __syncthreads();
    if (active) {
      __builtin_prefetch(arow + kc + 64, 0, 1);   // next A slab (global_prefetch_b8)
      #pragma unroll
      for (int kk = 0; kk < 64; kk += 4) {
        v2f a, b;
        a.x = arow[kc + kk];
        a.y = arow[kc + kk + 1];
        b.x = sW[l16][kk + 2 * half];
        b.y = sW[l16][kk + 2 * half + 1];
        acc = __builtin_amdgcn_wmma_f32_16x16x4_f32(false, a, false, b, (short)0, acc,
                                                    false, false);
      }
    }
  }

  if (!active) return;
  int n = ct * 16 + l16;
  if (n < Nout) {
    float bv = bias ? bias[n] : 0.f;
    #pragma unroll
    for (int r = 0; r < 8; ++r) {
      int m = rt * 16 + r + half * 8;             // C/D layout: VGPR r -> M=r (+8 hi half)
      size_t orow = (size_t)(m / rowsPerBatch) * outRowsPerBatch + (m % rowsPerBatch);
      float v = acc[r] + bv;
      if (act == 1) v = gelu_exact(v);
      if (res) v += res[orow * Nout + n];
      C[orow * Nout + n] = v;
    }
  }
}

// ---------------- dt = softplus(raw + dt_bias); decay = exp(dt * -exp(A_log)) ----

__global__ void dt_softplus(const float* __restrict__ zx, const float* __restrict__ dtbias,
                            const float* __restrict__ alog, float* __restrict__ dts,
                            float* __restrict__ dec, int total) {
  int i = blockIdx.x * blockDim.x + threadIdx.x;
  if (i >= total) return;
  int h = i & 7;
  int row = i >> 3;
  float x = zx[(size_t)row * ZROWS + 2176 + h] + dtbias[h];
  float sp = (x > 20.f) ? x : log1pf(expf(x));
  dts[i] = sp;
  dec[i] = expf(-sp * expf(alog[h]));
}

// ---------------- causal depthwise conv (width 4) + silu over xBC ----------------

__global__ void dconv_silu(const float* __restrict__ zx, const float* __restrict__ cw,
                           const float* __restrict__ cb, float* __restrict__ out,
                           int Lt, int total) {
  int i = blockIdx.x * blockDim.x + threadIdx.x;
  if (i >= total) return;
  int c = i % CONVD;
  int l = (i / CONVD) % Lt;
  int b = i / (CONVD * Lt);
  float acc = cb[c];
  #pragma unroll
  for (int j = 0; j < 4; ++j) {
    int tl = l - 3 + j;
    if (tl >= 0)
      acc += zx[((size_t)(b * Lt + tl)) * ZROWS + 1024 + c] * cw[c * 4 + j];
  }
  out[((size_t)(b * Lt + l)) * CONVD + c] = silu(acc);
}

// ---------------- sequential SSM scan: one workgroup per (b,h) ----------------
// thread t owns p = t>>1, n in [(t&1)*32, +32); 32 f32 state regs per thread.

__global__ void ssm_scan(const float* __restrict__ xbcc, const float* __restrict__ dts,
                         const float* __restrict__ dec, float* __restrict__ ys, int Lt) {
  const int h = blockIdx.x, b = blockIdx.y;
  const int tid = threadIdx.x;
  const int p = tid >> 1;
  const int n0 = (tid & 1) * 32;
  float st[32];
  #pragma unroll
  for (int i = 0; i < 32; ++i) st[i] = 0.f;
  __shared__ float sB[64], sC[64], sMeta[2];

  for (int t = 0; t < Lt; ++t) {
    size_t row = (size_t)b * Lt + t;
    if (tid < 64)        sB[tid]       = xbcc[row * CONVD + 1024 + tid];
    else if (tid < 128)  sC[tid - 64]  = xbcc[row * CONVD + 1088 + (tid - 64)];
    else if (tid == 128) sMeta[0] = dec[row * 8 + h];
    else if (tid == 129) sMeta[1] = dts[row * 8 + h];
    __syncthreads();
    float dcy = sMeta[0];
    float dtx = sMeta[1] * xbcc[row * CONVD + h * PDIM + p];
    float part = 0.f;
    #pragma unroll
    for (int i = 0; i < 32; ++i) {
      st[i] = dcy * st[i] + dtx * sB[n0 + i];
      part += st[i] * sC[n0 + i];
    }
    part += __shfl_xor(part, 1, 32);              // pair-sum over the two n-halves
    if ((tid & 1) == 0) ys[row * DINNER + h * PDIM + p] = part;
    __syncthreads();
  }
}

// ---------------- y = (ys + D*xs) * silu(z); RMS-norm * norm_w (in place) -----

__global__ void gate_rmsnorm(const float* __restrict__ xbcc, const float* __restrict__ zx,
                             const float* __restrict__ Dp, const float* __restrict__ nw,
                             float* __restrict__ ys, int Lt) {
  __shared__ float red[256];
  size_t row = (size_t)blockIdx.y * Lt + blockIdx.x;
  float vals[4];
  float ss = 0.f;
  #pragma unroll
  for (int j = 0; j < 4; ++j) {
    int i = threadIdx.x + j * 256;
    float y = ys[row * DINNER + i] + Dp[i >> 7] * xbcc[row * CONVD + i];
    float z = zx[row * ZROWS + i];
    y *= silu(z);
    vals[j] = y;
    ss += y * y;
  }
  float ms = blockReduceSum256(ss, red) * (1.f / 1024.f);
  float rs = rsqrtf(ms + LN_EPS);
  #pragma unroll
  for (int j = 0; j < 4; ++j) {
    int i = threadIdx.x + j * 256;
    ys[row * DINNER + i] = vals[j] * rs * nw[i];
  }
}

// ---------------- slice last Lloc, + local_norm residual, layernorm (ln3) -----

__global__ void add_ln3(const float* __restrict__ mo, const float* __restrict__ xcat,
                        const float* __restrict__ g, const float* __restrict__ bb,
                        float* __restrict__ out, int Lt, int Lloc) {
  __shared__ float red[256];
  int l = blockIdx.x, b = blockIdx.y, t = threadIdx.x;
  size_t row = (size_t)b * Lt + (size_t)(Lt - Lloc) + l;
  float x0 = mo[row * DM + t]       + xcat[row * DM + t];
  float x1 = mo[row * DM + t + 256] + xcat[row * DM + t + 256];
  float mu = blockReduceSum256(x0 + x1, red) * (1.f / 512.f);
  float d0 = x0 - mu, d1 = x1 - mu;
  float var = blockReduceSum256(d0 * d0 + d1 * d1, red) * (1.f / 512.f);
  float rs = rsqrtf(var + LN_EPS);
  float* o = out + ((size_t)(b * Lloc + l)) * DM;
  o[t]       = d0 * rs * g[t]       + bb[t];
  o[t + 256] = d1 * rs * g[t + 256] + bb[t + 256];
}

// ---------------------------------------------------------------------------

extern "C" void kernel_launch(void* const* d_in, const int* in_sizes, int n_in,
                              void* d_out, int out_size, void* d_ws, size_t ws_size,
                              hipStream_t stream) {
  (void)in_sizes; (void)n_in; (void)out_size; (void)ws_size;

  // setup_inputs() flattening order:
  const float* local_repr  = (const float*)d_in[0];   // (4,4096,512)
  const float* global_repr = (const float*)d_in[1];   // (4,512,512)
  const float* x_extra     = (const float*)d_in[2];   // (4,256)
  const float* ln1_g = (const float*)d_in[3];
  const float* ln1_b = (const float*)d_in[4];
  const float* ln2_g = (const float*)d_in[5];
  const float* ln2_b = (const float*)d_in[6];
  const float* ln3_g = (const float*)d_in[7];
  const float* ln3_b = (const float*)d_in[8];
  const float* emb_w    = (const float*)d_in[9];
  const float* emb_ln_g = (const float*)d_in[10];
  const float* emb_ln_b = (const float*)d_in[11];
  // emb_mamba: Win, conv_w, conv_b, dt_bias, A_log, D, norm_w, Wout
  const float* eWin  = (const float*)d_in[12];
  const float* eCw   = (const float*)d_in[13];
  const float* eCb   = (const float*)d_in[14];
  const float* eDtb  = (const float*)d_in[15];
  const float* eAlog = (const float*)d_in[16];
  const float* eD    = (const float*)d_in[17];
  const float* eNw   = (const float*)d_in[18];
  const float* eWout = (const float*)d_in[19];
  // ssd mamba
  const float* sWin  = (const float*)d_in[20];
  const float* sCw   = (const float*)d_in[21];
  const float* sCb   = (const float*)d_in[22];
  const float* sDtb  = (const float*)d_in[23];
  const float* sAlog = (const float*)d_in[24];
  const float* sD    = (const float*)d_in[25];
  const float* sNw   = (const float*)d_in[26];
  const float* sWout = (const float*)d_in[27];
  const float* W1 = (const float*)d_in[28];
  const float* b1 = (const float*)d_in[29];
  const float* W2 = (const float*)d_in[30];
  const float* b2 = (const float*)d_in[31];
  // d_in[32]/[33] = start/end offsets (0 in this configuration)

  const int B = 4, Lloc = 4096, G = 512, E = 256;
  const int Lt  = E + G + Lloc;          // 4864
  const int Rss = B * Lt;                // 19456 (mult of 16)
  const int Re  = B * E;                 // 1024
  const int Rff = B * Lloc;              // 16384

  // workspace layout (f32 elements)
  float* ws = (float*)d_ws;
  size_t off = 0;
  auto alloc = [&](size_t n) { float* p = ws + off; off += n; return p; };
  float* xcat  = alloc((size_t)Rss * DM);      // concat input, holds global/local norms
  float* zx    = alloc((size_t)Rss * ZROWS);   // z | xBC_raw | dt
  float* xbcc  = alloc((size_t)Rss * CONVD);   // conv+silu output (xs|B|C)
  float* ysb   = alloc((size_t)Rss * DINNER);  // scan out, then gated+rmsnormed y
  float* dtb   = alloc((size_t)Rss * NHEADS);
  float* decb  = alloc((size_t)Rss * NHEADS);
  float* mo    = alloc((size_t)Rss * DM);      // ssd mamba output (full seq)
  float* xln   = alloc((size_t)Rff * DM);      // post-residual LN3
  float* hid   = alloc((size_t)Rff * 256);     // FF hidden
  float* hb    = alloc((size_t)Re * DM);       // emb h
  float* zxe   = alloc((size_t)Re * ZROWS);
  float* xbcce = alloc((size_t)Re * CONVD);
  float* yse   = alloc((size_t)Re * DINNER);
  float* dtbe  = alloc((size_t)Re * NHEADS);
  float* decbe = alloc((size_t)Re * NHEADS);

  auto gemm = [&](const float* A, const float* W, const float* bias, const float* res,
                  float* C, int R, int K, int Nout, int act, int rpb, int orpb) {
    dim3 grid((Nout + 15) / 16, (R / 16 + 7) / 8);
    gemm_wmma_f32<<<grid, 256, 0, stream>>>(A, W, bias, res, C, R, K, Nout, act, rpb, orpb);
  };

  // --- build concat buffer: [extra_token | global_norm | local_norm] ---
  ln_rows<<<dim3(Lloc, B), 256, 0, stream>>>(local_repr, ln1_g, ln1_b,
      xcat + (size_t)(E + G) * DM, (size_t)Lloc * DM, (size_t)Lt * DM);
  ln_rows<<<dim3(G, B), 256, 0, stream>>>(global_repr, ln2_g, ln2_b,
      xcat + (size_t)E * DM, (size_t)G * DM, (size_t)Lt * DM);
  emb_h_ln<<<dim3(E, B), 256, 0, stream>>>(x_extra, emb_w, emb_ln_g, emb_ln_b, hb, E);

  // --- emb mamba on (B, E, DM) -> writes rows [0,E) of xcat per batch ---
  gemm(hb, eWin, nullptr, nullptr, zxe, Re, DM, ZROWS, 0, Re, Re);
  dt_softplus<<<(Re * NHEADS + 255) / 256, 256, 0, stream>>>(zxe, eDtb, eAlog, dtbe, decbe, Re * NHEADS);
  dconv_silu<<<((size_t)Re * CONVD + 255) / 256, 256, 0, stream>>>(zxe, eCw, eCb, xbcce, E, Re * CONVD);
  ssm_scan<<<dim3(NHEADS, B), 256, 0, stream>>>(xbcce, dtbe, decbe, yse, E);
  gate_rmsnorm<<<dim3(E, B), 256, 0, stream>>>(xbcce, zxe, eD, eNw, yse, E);
  gemm(yse, eWout, nullptr, nullptr, xcat, Re, DINNER, DM, 0, E, Lt);  // batch-strided out

  // --- ssd mamba on (B, Lt, DM) ---
  gemm(xcat, sWin, nullptr, nullptr, zx, Rss, DM, ZROWS, 0, Rss, Rss);
  dt_softplus<<<(Rss * NHEADS + 255) / 256, 256, 0, stream>>>(zx, sDtb, sAlog, dtb, decb, Rss * NHEADS);
  dconv_silu<<<((size_t)Rss * CONVD + 255) / 256, 256, 0, stream>>>(zx, sCw, sCb, xbcc, Lt, Rss * CONVD);
  ssm_scan<<<dim3(NHEADS, B), 256, 0, stream>>>(xbcc, dtb, decb, ysb, Lt);
  gate_rmsnorm<<<dim3(Lt, B), 256, 0, stream>>>(xbcc, zx, sD, sNw, ysb, Lt);
  gemm(ysb, sWout, nullptr, nullptr, mo, Rss, DINNER, DM, 0, Rss, Rss);

  // --- slice last Lloc, residual with local_norm (same rows of xcat), LN3 ---
  add_ln3<<<dim3(Lloc, B), 256, 0, stream>>>(mo, xcat, ln3_g, ln3_b, xln, Lt, Lloc);

  // --- FF: gelu(x@W1^T + b1) @ W2^T + b2 + x -> d_out ---
  gemm(xln, W1, b1, nullptr, hid, Rff, DM, 256, 1, Rff, Rff);
  gemm(hid, W2, b2, xln, (float*)d_out, Rff, 256, DM, 0, Rff, Rff);
}